// StandardRasterizer_51307679318773
// MI455X (gfx1250) — compile-verified
//
#include <hip/hip_runtime.h>
#include <hip/hip_bf16.h>

typedef __attribute__((ext_vector_type(2))) float v2f;
typedef __attribute__((ext_vector_type(8))) float v8f;

#define HH 256
#define WW 256
#define NPIX (HH * WW)
#define F_TRI 9976
#define NT_PAD 9984              // next multiple of 16
#define NGROUP (NT_PAD / 16)     // 624 groups of 16 triangles
#define BIGF 1000000.0f
#define EPSF 1e-12f

// ---------------------------------------------------------------------------
// Kernel 1: per-triangle plane setup.
// Affine planes w(gx,gy)=A*gx+B*gy+C for w0,w1,w2 and inv_z, stored in
// V_WMMA_F32_16X16X4_F32 A-fragment lane order:
//   lane n   (n<16): (K=0,K=1) = (A,B) of triangle n
//   lane n+16      : (K=2,K=3) = (C,0) of triangle n
// ws layout: [group][plane 0..3][lane 0..31][2 floats] -> 1KB per group.
// ---------------------------------------------------------------------------
__global__ __launch_bounds__(256) void setup_planes(const float* __restrict__ fv,
                                                    float* __restrict__ ws) {
  int i = blockIdx.x * blockDim.x + threadIdx.x;
  if (i >= NT_PAD) return;
  float p[4][3];
  if (i < F_TRI) {
    const float* t = fv + (size_t)i * 9;
    float x0 = t[0], y0 = t[1], z0 = t[2];
    float x1 = t[3], y1 = t[4], z1 = t[5];
    float x2 = t[6], y2 = t[7], z2 = t[8];
    float d = (y1 - y2) * (x0 - x2) + (x2 - x1) * (y0 - y2);
    if (fabsf(d) < EPSF) d = EPSF;
    float A0 = (y1 - y2) / d, B0 = (x2 - x1) / d;
    float C0 = -(A0 * x2 + B0 * y2);
    float A1 = (y2 - y0) / d, B1 = (x0 - x2) / d;
    float C1 = -(A1 * x2 + B1 * y2);
    float A2 = -A0 - A1, B2 = -B0 - B1, C2 = 1.0f - C0 - C1;
    float iz0 = 1.0f / z0, iz1 = 1.0f / z1, iz2 = 1.0f / z2;
    p[0][0] = A0; p[0][1] = B0; p[0][2] = C0;
    p[1][0] = A1; p[1][1] = B1; p[1][2] = C1;
    p[2][0] = A2; p[2][1] = B2; p[2][2] = C2;
    p[3][0] = A0 * iz0 + A1 * iz1 + A2 * iz2;
    p[3][1] = B0 * iz0 + B1 * iz1 + B2 * iz2;
    p[3][2] = C0 * iz0 + C1 * iz1 + C2 * iz2;
  } else {
    for (int q = 0; q < 4; ++q) { p[q][0] = 0.f; p[q][1] = 0.f; p[q][2] = 0.f; }
    p[0][2] = -1.0f;   // padding: w0 == -1 everywhere -> never inside
  }
  int g = i >> 4, n = i & 15;
  for (int q = 0; q < 4; ++q) {
    float* base = ws + (size_t)(g * 4 + q) * 64;
    base[n * 2 + 0]        = p[q][0];
    base[n * 2 + 1]        = p[q][1];
    base[(n + 16) * 2 + 0] = p[q][2];
    base[(n + 16) * 2 + 1] = 0.0f;
  }
}

// ---------------------------------------------------------------------------
// Kernel 2: WMMA rasterizer, triple-buffered depth-2 async pipeline.
// Block = 256 threads = 8 waves; each wave owns 16 pixels.  The block stages
// each 1KB triangle-group into LDS once via global_load_async_to_lds_b32.
// Pipeline depth 2: iteration g issues the prefetch of group g+2 and waits
// ASYNCcnt <= 1, draining only the prefetch of group g+1 (issued a whole
// iteration earlier) while group g+2 remains in flight across the barrier.
// ---------------------------------------------------------------------------
__global__ __launch_bounds__(256) void raster(const float* __restrict__ planes,
                                              int* __restrict__ tri_out) {
  __shared__ __align__(16) float buf[3][256];   // 3 x 1KB rotating buffers

  int tid = threadIdx.x;
  int gtid = blockIdx.x * 256 + tid;
  int wave = gtid >> 5;
  int lane = tid & 31;
  int pixBase = wave * 16;            // 16 consecutive row-major pixels
  int n = lane & 15;
  int pix = pixBase + n;
  float gx = (float)(pix & (WW - 1)) + 0.5f;
  float gy = (float)(pix >> 8) + 0.5f;

  // B fragment (4x16): K=0 row = gx, K=1 = gy, K=2 = 1, K=3 = 0
  v2f bfrag;
  bfrag.x = (lane < 16) ? gx : 1.0f;
  bfrag.y = (lane < 16) ? gy : 0.0f;

  float best_iz = 0.0f;               // valid inv_z >= 1/101 > 0
  int best_id = -1;
  const int idBase = (lane < 16) ? 0 : 8;
  const v8f zeroc = {};

#define PREFETCH(grp, slot)                                                    \
  {                                                                            \
    unsigned lds_off = (unsigned)(size_t)&buf[(slot)][tid];                    \
    unsigned goff = (unsigned)(((grp) * 256 + tid) * 4);                       \
    asm volatile("global_load_async_to_lds_b32 %0, %1, %2"                     \
                 :: "v"(lds_off), "v"(goff), "s"(planes) : "memory");          \
  }

  // Prologue: groups 0 and 1 in flight; drain group 0, keep group 1 flying.
  PREFETCH(0, 0)
  PREFETCH(1, 1)
  asm volatile("s_wait_asynccnt 0x1" ::: "memory");
  __syncthreads();

  int sc = 0;   // slot holding current group g
  int sp = 2;   // slot receiving group g+2
  for (int g = 0; g < NGROUP; ++g) {
    // Consume group g from LDS (ds_load_b64, conflict-free).
    const v2f* ap = (const v2f*)buf[sc];
    v2f a0 = ap[0 * 32 + lane];
    v2f a1 = ap[1 * 32 + lane];
    v2f a2 = ap[2 * 32 + lane];
    v2f az = ap[3 * 32 + lane];
    v8f d0 = __builtin_amdgcn_wmma_f32_16x16x4_f32(false, a0, false, bfrag,
                                                   (short)0, zeroc, false, false);
    v8f d1 = __builtin_amdgcn_wmma_f32_16x16x4_f32(false, a1, false, bfrag,
                                                   (short)0, zeroc, false, false);
    v8f d2 = __builtin_amdgcn_wmma_f32_16x16x4_f32(false, a2, false, bfrag,
                                                   (short)0, zeroc, false, false);
    v8f dz = __builtin_amdgcn_wmma_f32_16x16x4_f32(false, az, false, bfrag,
                                                   (short)0, zeroc, false, false);

    // Prefetch group g+2 into slot (g+2)%3.  Its previous readers (iter g-1)
    // are past the dscnt-wait + barrier that ended iter g-1, so this is safe
    // anywhere in the body.  Tail: redundant reload of the last group into a
    // slot that is never read again (distinct mod 3 from both live slots).
    int pg = g + 2 < NGROUP ? g + 2 : NGROUP - 1;
    PREFETCH(pg, sp)

#pragma unroll
    for (int r = 0; r < 8; ++r) {
      float m = fminf(d0[r], fminf(d1[r], d2[r]));
      float score = (m >= 0.0f) ? dz[r] : -1.0f;    // inside <=> min(w) >= 0
      bool better = score > best_iz;                // strict: first max wins
      int cand = g * 16 + idBase + r;
      best_iz = better ? score : best_iz;
      best_id = better ? cand : best_id;
    }

    // Drain only the OLDER prefetch (group g+1, in-order completion); group
    // g+2 stays in flight across the barrier.  __syncthreads supplies the
    // dscnt-wait (reads of slot sc drained) + block barrier.
    asm volatile("s_wait_asynccnt 0x1" ::: "memory");
    __syncthreads();

    sc = (sc == 2) ? 0 : sc + 1;
    sp = (sp == 2) ? 0 : sp + 1;
  }
#undef PREFETCH

  // Merge the two lane halves (triangles 0..7 vs 8..15 of each group);
  // on an exact tie prefer the lower triangle id (reference argmin semantics)
  float o_iz = __shfl_xor(best_iz, 16, 32);
  int o_id = __shfl_xor(best_id, 16, 32);
  bool take = (o_iz > best_iz) ||
              ((o_iz == best_iz) && (o_id >= 0) &&
               ((best_id < 0) || (o_id < best_id)));
  if (take) { best_iz = o_iz; best_id = o_id; }
  if (lane < 16) tri_out[pix] = best_id;
}

// ---------------------------------------------------------------------------
// Kernel 3: shading (exact reference math per winning pixel).
// ---------------------------------------------------------------------------
__global__ __launch_bounds__(256) void shade(const float* __restrict__ fv,
                                             const float* __restrict__ fc,
                                             const int* __restrict__ tri,
                                             float* __restrict__ images,
                                             float* __restrict__ depth) {
  int pix = blockIdx.x * blockDim.x + threadIdx.x;
  if (pix >= NPIX) return;
  int t = tri[pix];
  if (t < 0) {
    images[0 * NPIX + pix] = 0.0f;
    images[1 * NPIX + pix] = 0.0f;
    images[2 * NPIX + pix] = 0.0f;
    depth[pix] = BIGF;
    return;
  }
  const float* v = fv + (size_t)t * 9;
  float gx = (float)(pix & (WW - 1)) + 0.5f;
  float gy = (float)(pix >> 8) + 0.5f;
  float x0 = v[0], y0 = v[1], z0 = v[2];
  float x1 = v[3], y1 = v[4], z1 = v[5];
  float x2 = v[6], y2 = v[7], z2 = v[8];
  float d = (y1 - y2) * (x0 - x2) + (x2 - x1) * (y0 - y2);
  if (fabsf(d) < EPSF) d = EPSF;
  float w0 = ((y1 - y2) * (gx - x2) + (x2 - x1) * (gy - y2)) / d;
  float w1 = ((y2 - y0) * (gx - x2) + (x0 - x2) * (gy - y2)) / d;
  float w2 = 1.0f - w0 - w1;
  float wc0 = w0 / z0, wc1 = w1 / z1, wc2 = w2 / z2;
  float s = wc0 + wc1 + wc2;
  wc0 /= s; wc1 /= s; wc2 /= s;
  const float* c = fc + (size_t)t * 9;
#pragma unroll
  for (int ch = 0; ch < 3; ++ch) {
    images[ch * NPIX + pix] =
        wc0 * c[0 * 3 + ch] + wc1 * c[1 * 3 + ch] + wc2 * c[2 * 3 + ch];
  }
  depth[pix] = wc0 * z0 + wc1 * z1 + wc2 * z2;
}

// ---------------------------------------------------------------------------
// d_in[0] = face_vertices (1,9976,3,3) f32
// d_in[1] = face_colors   (1,9976,3,3) f32
// d_in[2] = return_buffers (scalar, ignored; always write all three outputs)
// d_out   = [images 3*65536 f32][tri 65536 i32][depth 65536 f32]
// d_ws    = plane table, 624*4*64 floats = 638,976 bytes
// ---------------------------------------------------------------------------
extern "C" void kernel_launch(void* const* d_in, const int* in_sizes, int n_in,
                              void* d_out, int out_size, void* d_ws, size_t ws_size,
                              hipStream_t stream) {
  (void)in_sizes; (void)n_in; (void)out_size; (void)ws_size;
  const float* fv = (const float*)d_in[0];
  const float* fc = (const float*)d_in[1];
  float* out = (float*)d_out;
  float* images = out;                      // 3*65536
  int* tri = (int*)(out + 3 * NPIX);        // 65536 (int32 bits)
  float* depth = out + 4 * NPIX;            // 65536
  float* planes = (float*)d_ws;

  setup_planes<<<NT_PAD / 256, 256, 0, stream>>>(fv, planes);

  int raster_threads = (NPIX / 16) * 32;    // 16 pixels per wave32
  raster<<<raster_threads / 256, 256, 0, stream>>>(planes, tri);

  shade<<<NPIX / 256, 256, 0, stream>>>(fv, fc, tri, images, depth);
}